// SpikeFP32Softmax_84713934946544
// MI455X (gfx1250) — compile-verified
//
#include <hip/hip_runtime.h>

// SpikeFP32Softmax: decode 32-pulse FP32 words, raw softmax (exp/sum/div, no
// max subtraction) over N=256, re-encode to pulses.
// Input  x  : [8, 2048, 256, 32] float32 in {0,1}, MSB-first IEEE-754 bits
// Output out: same shape/encoding.
//
// HBM-bound: 1 GiB traffic @ 23.3 TB/s -> ~46 us floor. CDNA5 async
// global<->LDS DMA streams coalesced 512 B/wave-op both directions; padded
// LDS layout (144 B/element) makes the per-element bit gather/scatter
// bank-conflict-optimal; decode is an exact split FMA dot product (1 VALU
// op per bit); wave32 shfl tree + LDS for the N=256 reduction.

#define NT          256                  // threads per block == elements per row
#define BITS        32
#define CHUNKS      8                    // 8 x float4 per element
#define PAD_CHUNKS  9                    // padded element stride in float4 (144 B)
#define ROW_FLOATS  (NT * BITS)          // 8192 floats = 32 KB per row
#define LDS_FLOAT4  (NT * PAD_CHUNKS)    // 2304 float4 = 36 KB staging

// weight of bit k (MSB-first): 2^(31-k), exact compile-time float
#define W(k) ((float)(1u << (31 - (k))))

// ---------------- CDNA5 async DMA helpers (gfx1250) ----------------
__device__ __forceinline__ unsigned lds_off_u32(const void* p) {
#if defined(__HIP_DEVICE_COMPILE__)
  // generic -> LDS addrspace pointer; its 32-bit value is the LDS byte offset
  return (unsigned)(unsigned long long)(__attribute__((address_space(3))) const void*)p;
#else
  (void)p; return 0u;
#endif
}

__device__ __forceinline__ void async_load_b128(unsigned lds_off, const void* gsrc) {
#if defined(__HIP_DEVICE_COMPILE__)
  asm volatile("global_load_async_to_lds_b128 %0, %1, off"
               :: "v"(lds_off), "v"((unsigned long long)(size_t)gsrc)
               : "memory");
#else
  (void)lds_off; (void)gsrc;
#endif
}

__device__ __forceinline__ void async_store_b128(void* gdst, unsigned lds_off) {
#if defined(__HIP_DEVICE_COMPILE__)
  asm volatile("global_store_async_from_lds_b128 %0, %1, off"
               :: "v"((unsigned long long)(size_t)gdst), "v"(lds_off)
               : "memory");
#else
  (void)gdst; (void)lds_off;
#endif
}

__device__ __forceinline__ void wait_asynccnt0() {
#if defined(__HIP_DEVICE_COMPILE__)
#if __has_builtin(__builtin_amdgcn_s_wait_asynccnt)
  __builtin_amdgcn_s_wait_asynccnt(0);
#else
  asm volatile("s_wait_asynccnt 0x0" ::: "memory");
#endif
#endif
}

// accumulate pulse b with bit index K into the right exact partial sum
#define ACCBIT(K, b)                              \
  do {                                            \
    if ((K) <= 8)       hi  = fmaf((b), W(K), hi);  \
    else if ((K) <= 20) loA = fmaf((b), W(K), loA); \
    else                loB = fmaf((b), W(K), loB); \
  } while (0)

// ---------------- kernel ----------------
__global__ __launch_bounds__(NT) void spike_fp32_softmax_kernel(
    const float* __restrict__ x, float* __restrict__ out) {
  __shared__ __align__(16) float4 buf[LDS_FLOAT4];  // 36 KB padded row staging
  __shared__ float wsum[NT / 32];                   // per-wave partial sums

  const int t = threadIdx.x;
  const long long row = blockIdx.x;
  const float* rin  = x   + row * (long long)ROW_FLOATS;
  float*       rout = out + row * (long long)ROW_FLOATS;

  const unsigned buf_base = lds_off_u32(buf);

  // ---- Phase A: async DMA global -> LDS (coalesced 512 B / wave op),
  //      scattered into padded layout: global chunk c -> LDS chunk c + c/8
#pragma unroll
  for (int i = 0; i < CHUNKS; ++i) {
    const int c = i * NT + t;          // linear float4 chunk within row
    const int p = c + (c >> 3);        // padded LDS float4 index
    async_load_b128(buf_base + (unsigned)p * 16u,
                    (const char*)rin + (size_t)c * 16u);
  }
  wait_asynccnt0();
  __syncthreads();   // element t's chunks were staged by other threads

  // ---- Phase B: decode element t via exact split-FMA dot product.
  //      bits 0..8 -> hi (multiples of 2^23, <=9 significand bits, exact)
  //      bits 9..31 -> loA/loB (integers < 2^23, exact)
  const float4* bv = (const float4*)buf;
  float hi = 0.0f, loA = 0.0f, loB = 0.0f;
#pragma unroll
  for (int j = 0; j < CHUNKS; ++j) {
    const float4 b = bv[t * PAD_CHUNKS + j];
    ACCBIT(4 * j + 0, b.x);
    ACCBIT(4 * j + 1, b.y);
    ACCBIT(4 * j + 2, b.z);
    ACCBIT(4 * j + 3, b.w);
  }
  const unsigned u = (unsigned)hi + (unsigned)(loA + loB);  // exact reassembly
  const float xf = __uint_as_float(u);
  const float e  = __expf(xf);         // raw exp, no max subtraction (as reference)

  // ---- Phase C: block-wide sum over N=256 (wave32 shfl tree + LDS cross-wave)
  float s = e;
#pragma unroll
  for (int off = 16; off > 0; off >>= 1) s += __shfl_xor(s, off, 32);
  if ((t & 31) == 0) wsum[t >> 5] = s;
  __syncthreads();
  float tot = 0.0f;
#pragma unroll
  for (int w = 0; w < NT / 32; ++w) tot += wsum[w];

#if defined(__HIP_DEVICE_COMPILE__) && __has_builtin(__builtin_amdgcn_rcpf)
  const float r = e * __builtin_amdgcn_rcpf(tot);
#else
  const float r = e / tot;
#endif

  // ---- Phase D: re-encode into own padded LDS region (thread-private)
  const unsigned v = __float_as_uint(r);
  float4* bw = (float4*)buf;
#pragma unroll
  for (int j = 0; j < CHUNKS; ++j) {
    const int k = 4 * j;
    float4 o;
    o.x = (float)((v >> (31 - k)) & 1u);
    o.y = (float)((v >> (30 - k)) & 1u);
    o.z = (float)((v >> (29 - k)) & 1u);
    o.w = (float)((v >> (28 - k)) & 1u);
    bw[t * PAD_CHUNKS + j] = o;
  }
  __syncthreads();   // all encode writes visible before DMA-out

  // ---- Phase E: async DMA LDS -> global (coalesced), gathering from padding
#pragma unroll
  for (int i = 0; i < CHUNKS; ++i) {
    const int c = i * NT + t;
    const int p = c + (c >> 3);
    async_store_b128((char*)rout + (size_t)c * 16u,
                     buf_base + (unsigned)p * 16u);
  }
  wait_asynccnt0();
}

// ---------------- launch ----------------
extern "C" void kernel_launch(void* const* d_in, const int* in_sizes, int n_in,
                              void* d_out, int out_size, void* d_ws, size_t ws_size,
                              hipStream_t stream) {
  (void)n_in; (void)out_size; (void)d_ws; (void)ws_size;
  const float* x = (const float*)d_in[0];
  float* out = (float*)d_out;

  const int rows = in_sizes[0] / ROW_FLOATS;   // 8*2048 = 16384
  dim3 grid((unsigned)rows), block(NT);
  hipLaunchKernelGGL(spike_fp32_softmax_kernel, grid, block, 0, stream, x, out);
}